// E_Recomposition_6073083756936
// MI455X (gfx1250) — compile-verified
//
#include <hip/hip_runtime.h>
#include <stdint.h>

#define N_ATOMS 1024
#define NC2     523776      // N*(N-1)/2, divisible by 16
#define BATCH   32
#define CHUNKS  128
#define PAIRS_PER_CHUNK 4096 // CHUNKS*PAIRS_PER_CHUNK = 524288 >= NC2
#define TPB     256
#define RUN     16           // contiguous pairs per thread

// ---------------------------------------------------------------------------
// Pass 1: per-(batch, chunk) partial reductions.
//   s1 = sum 1/r^2,   s2 = sum (1/r) * decompFE
// coords[b] (12 KB packed xyz) is staged into LDS with CDNA5 async
// global->LDS copies (ASYNCcnt path), then expanded to a 16B-padded float4
// array so the hot loop does ONE aligned ds_load_b128 per atom access.
// The VMEM pipe is left dedicated to the 67 MB decompFE float4 stream.
// ---------------------------------------------------------------------------
__global__ __launch_bounds__(TPB) void pair_partials_kernel(
    const float* __restrict__ coords,   // [B][N][3]
    const float* __restrict__ decomp,   // [B][NC2]
    float2* __restrict__ partials)      // [B][CHUNKS]
{
    __shared__ __align__(16) float  stage[3 * N_ATOMS];  // 12 KB packed xyz
    __shared__ __align__(16) float4 scp[N_ATOMS];        // 16 KB padded xyz_
    __shared__ float red1[TPB];
    __shared__ float red2[TPB];

    const int chunk = blockIdx.x;
    const int b     = blockIdx.y;
    const int tid   = threadIdx.x;

    // ---- async stage packed coords[b] into LDS: 768 x 16B transfers ----
    {
        const float* gsrc = coords + (size_t)b * (3 * N_ATOMS);
        for (int t = tid; t < (3 * N_ATOMS) / 4; t += TPB) {
            uint32_t loff  = (uint32_t)(uintptr_t)(&stage[4 * t]); // low 32b of flat LDS addr == LDS offset
            const float* g = gsrc + 4 * t;
            asm volatile("global_load_async_to_lds_b128 %0, %1, off"
                         :: "v"(loff), "v"(g) : "memory");
        }
        asm volatile("s_wait_asynccnt 0" ::: "memory");
    }
    __syncthreads();

    // ---- expand packed [N][3] -> padded [N] float4 (one-time, per block) ----
    for (int a = tid; a < N_ATOMS; a += TPB)
        scp[a] = make_float4(stage[3 * a + 0], stage[3 * a + 1], stage[3 * a + 2], 0.0f);
    __syncthreads();

    float s1 = 0.0f, s2 = 0.0f;

    const int base = chunk * PAIRS_PER_CHUNK + tid * RUN;
    if (base < NC2) {   // NC2 % RUN == 0 -> a run is entirely valid or entirely out
        // Decode first (i, j) from linear strict-lower-triangle index:
        //   k = i*(i-1)/2 + j,  0 <= j < i
        unsigned k = (unsigned)base;
        float f = sqrtf((float)(8u * k + 1u));            // exact: 8k+1 < 2^24
        int i = (int)((1.0f + f) * 0.5f);
        while ((unsigned)(i * (i - 1) / 2) > k) --i;      // fix 1-ulp sqrt slop
        while ((unsigned)((i + 1) * i / 2) <= k) ++i;
        int j = (int)(k - (unsigned)(i * (i - 1) / 2));

        // Coalesced decompFE: 4 aligned float4 loads = one 64B line per lane.
        const float4* drow = (const float4*)(decomp + (size_t)b * NC2 + base);
        float4 d4[RUN / 4];
        #pragma unroll
        for (int q = 0; q < RUN / 4; ++q) d4[q] = drow[q];
        const float* dv = (const float*)d4;

        float4 ci = scp[i];
        #pragma unroll
        for (int p = 0; p < RUN; ++p) {
            float4 cj = scp[j];                           // one ds_load_b128
            float dx = ci.x - cj.x;
            float dy = ci.y - cj.y;
            float dz = ci.z - cj.z;
            float r2   = dx * dx + dy * dy + dz * dz;
            float rinv = __builtin_amdgcn_rsqf(r2);       // raw v_rsq_f32 (TRANS, co-executes)
            s1 += rinv * rinv;                            // == 1/r^2
            s2 += rinv * dv[p];
            // advance along the triangle
            ++j;
            if (j == i) {
                j = 0; ++i;
                if (p < RUN - 1)                          // avoid OOB reload on last step
                    ci = scp[i];
            }
        }
    }

    // ---- deterministic block tree reduction ----
    red1[tid] = s1;
    red2[tid] = s2;
    __syncthreads();
    #pragma unroll
    for (int s = TPB / 2; s > 0; s >>= 1) {
        if (tid < s) {
            red1[tid] += red1[tid + s];
            red2[tid] += red2[tid + s];
        }
        __syncthreads();
    }
    if (tid == 0)
        partials[(size_t)b * CHUNKS + chunk] = make_float2(red1[0], red2[0]);
}

// ---------------------------------------------------------------------------
// Pass 2: fold CHUNKS partials per batch, out[b] = S2 / sqrt(S1)
// ---------------------------------------------------------------------------
__global__ __launch_bounds__(CHUNKS) void finalize_kernel(
    const float2* __restrict__ partials,  // [B][CHUNKS]
    float* __restrict__ out)              // [B]
{
    __shared__ float r1[CHUNKS];
    __shared__ float r2[CHUNKS];
    const int b   = blockIdx.x;
    const int tid = threadIdx.x;

    float2 p = partials[(size_t)b * CHUNKS + tid];
    r1[tid] = p.x;
    r2[tid] = p.y;
    __syncthreads();
    #pragma unroll
    for (int s = CHUNKS / 2; s > 0; s >>= 1) {
        if (tid < s) {
            r1[tid] += r1[tid + s];
            r2[tid] += r2[tid + s];
        }
        __syncthreads();
    }
    if (tid == 0)
        out[b] = r2[0] / sqrtf(r1[0]);
}

extern "C" void kernel_launch(void* const* d_in, const int* in_sizes, int n_in,
                              void* d_out, int out_size, void* d_ws, size_t ws_size,
                              hipStream_t stream) {
    const float* coords = (const float*)d_in[0];   // [32][1024][3] fp32
    const float* decomp = (const float*)d_in[1];   // [32][523776] fp32
    float2* partials    = (float2*)d_ws;           // 32*128*8 B = 32 KB scratch

    pair_partials_kernel<<<dim3(CHUNKS, BATCH), TPB, 0, stream>>>(coords, decomp, partials);
    finalize_kernel<<<BATCH, CHUNKS, 0, stream>>>(partials, (float*)d_out);
}